// TSBarrierModel_12051678233119
// MI455X (gfx1250) — compile-verified
//
#include <hip/hip_runtime.h>
#include <math.h>

// CDNA5 / gfx1250: wave32, WMMA f32 16x16x4 (full fp32 precision path)
typedef __attribute__((ext_vector_type(2))) float v2f;
typedef __attribute__((ext_vector_type(8))) float v8f;

#define NUM_BASIS 10
#define HID 16
#define WAVES_PER_BLOCK 8
#define XSTRIDE 36   // padded LDS row stride (floats), keeps 16B alignment

// soft-one-hot radial basis value, branchless (mirrors reference's where()):
//   diff = (r - v_b)/step ; denom = 1 - diff^2
//   f = inside ? 1.14136*exp(2 - 2/denom)*sqrt(10) : 0
__device__ __forceinline__ float emb_basis(float r, int b, bool valid) {
    const float STEP     = 3.0f / 11.0f;
    const float INV_STEP = 11.0f / 3.0f;
    const float AMP      = 1.14136f * 3.1622776601683795f;  // 1.14136*sqrt(10)
    float vb    = (float)(b + 1) * STEP;
    float diff  = (r - vb) * INV_STEP;
    float denom = 1.0f - diff * diff;
    bool inside = (denom > 0.0f) & (b < NUM_BASIS) & valid;
    float safe  = inside ? denom : 1.0f;                    // v_cndmask, no branch
    // v_rcp_f32 instead of IEEE divide sequence (~1ulp, fine for exp() argument)
    float f = AMP * __expf(2.0f - 2.0f * __builtin_amdgcn_rcpf(safe));
    return inside ? f : 0.0f;                               // v_cndmask, no branch
}

// W2r[k][c] = 0.25 * sum_v W2[k][p*64 + u*8 + v],  c = p*8+u  (0.25 = 1/sqrt(HID))
__global__ void w2r_kernel(const float* __restrict__ W2, float* __restrict__ W2r) {
    int idx = threadIdx.x;
    if (idx < HID * 32) {
        int k = idx >> 5, c = idx & 31;
        int p = c >> 3, u = c & 7;
        const float* base = W2 + k * 256 + p * 64 + u * 8;
        float s = 0.0f;
#pragma unroll
        for (int v = 0; v < 8; ++v) s += base[v];
        W2r[idx] = 0.25f * s;
    }
}

__global__ __launch_bounds__(256)
void edge_kernel(const float* __restrict__ in_feat,
                 const float* __restrict__ edge_vec,
                 const float* __restrict__ W1,
                 const float* __restrict__ W2r,
                 const int*   __restrict__ edge_src,
                 int E,
                 float* __restrict__ block_partials) {
    __shared__ float lds_x[WAVES_PER_BLOCK][16][XSTRIDE];
    __shared__ float lds_c[WAVES_PER_BLOCK][16][XSTRIDE];
    __shared__ float red[WAVES_PER_BLOCK];

    const int lane = threadIdx.x & 31;
    const int w    = threadIdx.x >> 5;
    const int e16  = lane & 15;   // edge-in-tile (A-matrix row M) / B-matrix column N
    const int t    = lane >> 4;   // half-wave selects K sub-pair per ISA A/B layout

    const int tile  = blockIdx.x * WAVES_PER_BLOCK + w;
    const int ebase = tile * 16;
    const int eg    = ebase + e16;
    const bool valid = (eg < E);
    const int ec    = valid ? eg : (E - 1);

    const float INV_SQRT10 = 0.31622776601683794f;
    const float SQRT2      = 1.4142135623730951f;
    const float SQRT3      = 1.7320508075688772f;
    // coeff scales (neighbor 1/sqrt(16)=0.25 folded in):
    const float C0 = 0.0625f;          // pw0*0.25
    const float C1 = 0.03608439182f;   // pw0*(1/sqrt3)*0.25
    const float C2 = 0.0625f;          // pw1*(1/sqrt3)*0.25 == 0.25*0.25
    const float C3 = 0.0625f;

    // ---- edge-independent B fragments (W1/sqrt10: K=12 padded x N=16) ----
    v2f bW1[3];
#pragma unroll
    for (int kk = 0; kk < 3; ++kk) {
        int b0 = 4 * kk + 2 * t;
        bW1[kk].x = (b0     < NUM_BASIS) ? W1[(b0    ) * HID + e16] * INV_SQRT10 : 0.0f;
        bW1[kk].y = (b0 + 1 < NUM_BASIS) ? W1[(b0 + 1) * HID + e16] * INV_SQRT10 : 0.0f;
    }
    // ---- edge-independent B fragments (W2r^T: K=32 coeff x N=16 hid) ----
    v2f bW2[8];
#pragma unroll
    for (int kk = 0; kk < 8; ++kk) {
        const float* p = W2r + e16 * 32 + 4 * kk + 2 * t;
        bW2[kk].x = p[0];
        bW2[kk].y = p[1];
    }

    // ---- per-edge geometry (both half-waves compute for edge e16) ----
    float vx = edge_vec[ec * 3 + 0];
    float vy = edge_vec[ec * 3 + 1];
    float vz = edge_vec[ec * 3 + 2];
    float r    = sqrtf(vx * vx + vy * vy + vz * vz);
    float invr = __builtin_amdgcn_rcpf(r);     // v_rcp_f32, avoids IEEE div sequence
    float shx = SQRT3 * vx * invr, shy = SQRT3 * vy * invr, shz = SQRT3 * vz * invr;
    float shsum = shx + shy + shz;

    // ---- stage gathered node features: each lane loads 16 floats (half a row) ----
    int src = edge_src[ec];
    const float4* xrow = (const float4*)(in_feat + (size_t)src * 32 + 16 * t);
    float4 x0 = xrow[0], x1 = xrow[1], x2 = xrow[2], x3 = xrow[3];
    float4* lx = (float4*)&lds_x[w][e16][16 * t];
    lx[0] = x0; lx[1] = x1; lx[2] = x2; lx[3] = x3;
    __syncthreads();

    // ---- build coeff[32] per edge (t=0: paths 0,1 ; t=1: paths 2,3) ----
    const float* xe = lds_x[w][e16];
    float*       ce = lds_c[w][e16];
    const float vmask = valid ? 1.0f : 0.0f;   // branchless tail handling
    if (t == 0) {
#pragma unroll
        for (int u = 0; u < 8; ++u) {
            float s = xe[u];
            const float* vv = xe + 8 + 3 * u;
            float vdot = vv[0] * shx + vv[1] * shy + vv[2] * shz;
            ce[u]     = vmask * C0 * s;
            ce[8 + u] = vmask * C1 * vdot;
        }
    } else {
#pragma unroll
        for (int u = 0; u < 8; ++u) {
            float s = xe[u];
            const float* vv = xe + 8 + 3 * u;
            float vsum = vv[0] + vv[1] + vv[2];
            ce[16 + u] = vmask * C2 * shsum * s;
            ce[24 + u] = vmask * C3 * vsum;
        }
    }
    __syncthreads();

    // ---- WMMA #1: H[16e x 16hid] = emb[16 x 12pad] @ (W1/sqrt10) ----
    v8f hacc = {};
#pragma unroll
    for (int kk = 0; kk < 3; ++kk) {
        int b0 = 4 * kk + 2 * t;
        v2f a;
        a.x = emb_basis(r, b0,     valid);
        a.y = emb_basis(r, b0 + 1, valid);
        hacc = __builtin_amdgcn_wmma_f32_16x16x4_f32(
            false, a, false, bW1[kk], (short)0, hacc, false, false);
    }
#pragma unroll
    for (int j = 0; j < 8; ++j) hacc[j] = fmaxf(hacc[j], 0.0f) * SQRT2;  // relu*sqrt2

    // ---- WMMA #2: G[16e x 16hid] = coeff[16 x 32] @ W2r^T ----
    v8f gacc = {};
#pragma unroll
    for (int kk = 0; kk < 8; ++kk) {
        const float* p = &lds_c[w][e16][4 * kk + 2 * t];
        v2f a; a.x = p[0]; a.y = p[1];
        gacc = __builtin_amdgcn_wmma_f32_16x16x4_f32(
            false, a, false, bW2[kk], (short)0, gacc, false, false);
    }

    // ---- tile contribution: sum over all elements of H .* G (same C-layout) ----
    float partial = 0.0f;
#pragma unroll
    for (int j = 0; j < 8; ++j) partial += hacc[j] * gacc[j];
#pragma unroll
    for (int off = 16; off > 0; off >>= 1) partial += __shfl_xor(partial, off, 32);

    if (lane == 0) red[w] = partial;
    __syncthreads();
    if (threadIdx.x == 0) {
        float s = 0.0f;
#pragma unroll
        for (int i = 0; i < WAVES_PER_BLOCK; ++i) s += red[i];
        block_partials[blockIdx.x] = s;
    }
}

__global__ void reduce_kernel(const float* __restrict__ partials, int n,
                              float* __restrict__ out) {
    __shared__ float sm[256];
    float s = 0.0f;
    for (int i = threadIdx.x; i < n; i += 256) s += partials[i];
    sm[threadIdx.x] = s;
    __syncthreads();
    for (int stride = 128; stride > 0; stride >>= 1) {
        if ((int)threadIdx.x < stride) sm[threadIdx.x] += sm[threadIdx.x + stride];
        __syncthreads();
    }
    if (threadIdx.x == 0) out[0] = sm[0];
}

extern "C" void kernel_launch(void* const* d_in, const int* in_sizes, int n_in,
                              void* d_out, int out_size, void* d_ws, size_t ws_size,
                              hipStream_t stream) {
    const float* in_feat  = (const float*)d_in[0];  // [N,32]
    const float* edge_vec = (const float*)d_in[1];  // [E,3]
    const float* W1       = (const float*)d_in[2];  // [10,16]
    const float* W2       = (const float*)d_in[3];  // [16,256]
    const int*   edge_src = (const int*)d_in[4];    // [E]
    // d_in[5] = edge_dst, d_in[6] = num_nodes: unused — output is a global sum,
    // so segment_sum followed by .sum() is just a sum over all edges.
    (void)n_in; (void)ws_size;

    int E = in_sizes[1] / 3;
    float* out      = (float*)d_out;
    float* W2r      = (float*)d_ws;          // 512 floats
    float* partials = (float*)d_ws + 512;    // one per block

    w2r_kernel<<<dim3(1), dim3(512), 0, stream>>>(W2, W2r);

    int tiles  = (E + 15) / 16;
    int blocks = (tiles + WAVES_PER_BLOCK - 1) / WAVES_PER_BLOCK;
    edge_kernel<<<dim3(blocks), dim3(256), 0, stream>>>(
        in_feat, edge_vec, W1, W2r, edge_src, E, partials);

    reduce_kernel<<<dim3(1), dim3(256), 0, stream>>>(partials, blocks, out);
    (void)out_size;
}